// mLSTM_25220047962819
// MI455X (gfx1250) — compile-verified
//
#include <hip/hip_runtime.h>
#include <math.h>

#define HID 4096

typedef __attribute__((ext_vector_type(2))) float v2f;
typedef __attribute__((ext_vector_type(4))) float v4f;
typedef __attribute__((ext_vector_type(8))) float v8f;

// ---------------------------------------------------------------------------
// Kernel 1: six fused GEMVs (Wq,Wk,Wv,Wi,Wf,Wo) via V_WMMA_F32_16X16X4_F32.
//
// GEMV-on-WMMA trick (exact f32 precision):
//   A (16x4) = x-chunk broadcast to all 16 M rows
//   B (4x16) column n = weight row (r0+n) over the same K-chunk
//   => D[m][n] = dot(x[k0..k0+3], W[r0+n][k0..k0+3]) for every m,
//      accumulated over k in the f32 C/D accumulator.
//
// f32 A layout (ISA 7.12.2): lanes 0-15 hold K={0,1}, lanes 16-31 K={2,3};
// B mirrors, so each lane loads one aligned float2 of its weight row
// (pair offset po = lane>=16). D VGPR0, lanes 0-15 = row-n results.
//
// Concurrency: one block = one 16-row group; its 8 waves split K 8 ways
// (512 K each). 1536 blocks -> 12288 waves keep ~12-15 MB of reads in
// flight (BW x latency for 23.3 TB/s HBM). Partials combine via an 8x16
// LDS reduction.
// ---------------------------------------------------------------------------
__global__ __launch_bounds__(256)
void mlstm_gemv6(const float* __restrict__ x,
                 const float* __restrict__ W0, const float* __restrict__ B0,
                 const float* __restrict__ W1, const float* __restrict__ B1,
                 const float* __restrict__ W2, const float* __restrict__ B2,
                 const float* __restrict__ W3, const float* __restrict__ B3,
                 const float* __restrict__ W4, const float* __restrict__ B4,
                 const float* __restrict__ W5, const float* __restrict__ B5,
                 float* __restrict__ gates /* 6*HID */) {
  __shared__ __align__(16) float xs[HID];
  __shared__ float red[8][16];
  for (int t = threadIdx.x; t < HID / 4; t += 256)
    ((v4f*)xs)[t] = ((const v4f*)x)[t];
  __syncthreads();

  const int wave = threadIdx.x >> 5;         // K slice 0..7 (512 K each)
  const int lane = threadIdx.x & 31;
  const int j    = blockIdx.x >> 8;          // which gate matrix 0..5
  const int r0   = (blockIdx.x & 255) << 4;  // row base within matrix

  const float* W = W0; const float* B = B0;
  if      (j == 1) { W = W1; B = B1; }
  else if (j == 2) { W = W2; B = B2; }
  else if (j == 3) { W = W3; B = B3; }
  else if (j == 4) { W = W4; B = B4; }
  else if (j == 5) { W = W5; B = B5; }

  const int n  = lane & 15;
  const int po = lane >> 4;                  // pair offset: 0 => K{0,1}, 1 => K{2,3}
  const v2f* __restrict__ wrow =
      (const v2f*)(W + (size_t)(r0 + n) * HID) + wave * (HID / 16);
  const v2f* __restrict__ xv = (const v2f*)xs + wave * (HID / 16);

  v8f c0 = {}; v8f c1 = {}; v8f c2 = {}; v8f c3 = {};
  // kp = float-pair index within slice; 8 pairs (16 K) per iter -> 4 WMMAs
#pragma unroll 4
  for (int kp = 0; kp < HID / 16; kp += 8) {
    v2f b0 = wrow[kp + 0 + po];
    v2f b1 = wrow[kp + 2 + po];
    v2f b2 = wrow[kp + 4 + po];
    v2f b3 = wrow[kp + 6 + po];
    v2f a0 = xv[kp + 0 + po];
    v2f a1 = xv[kp + 2 + po];
    v2f a2 = xv[kp + 4 + po];
    v2f a3 = xv[kp + 6 + po];
    c0 = __builtin_amdgcn_wmma_f32_16x16x4_f32(false, a0, false, b0, (short)0, c0, false, false);
    c1 = __builtin_amdgcn_wmma_f32_16x16x4_f32(false, a1, false, b1, (short)0, c1, false, false);
    c2 = __builtin_amdgcn_wmma_f32_16x16x4_f32(false, a2, false, b2, (short)0, c2, false, false);
    c3 = __builtin_amdgcn_wmma_f32_16x16x4_f32(false, a3, false, b3, (short)0, c3, false, false);
  }
  // Only D element 0 (M=0 row) is needed; lanes 0-15 hold rows r0+0..15.
  float zpart = (c0[0] + c1[0]) + (c2[0] + c3[0]);
  if (lane < 16) red[wave][lane] = zpart;
  __syncthreads();

  if (wave == 0 && lane < 16) {
    const int row = r0 + lane;
    float z = (((red[0][lane] + red[1][lane]) + (red[2][lane] + red[3][lane])) +
               ((red[4][lane] + red[5][lane]) + (red[6][lane] + red[7][lane]))) +
              B[row];
    float g;
    if      (j == 0) g = z;                              // q
    else if (j == 1) g = z * (1.0f / 64.0f);             // k * H^-1/2
    else if (j == 2) g = z;                              // v
    else if (j == 3) g = expf(z);                        // i = exp
    else             g = 1.0f / (1.0f + expf(-z));       // f, o = sigmoid
    gates[j * HID + row] = g;
  }
}

// ---------------------------------------------------------------------------
// Kernel 2: n_t = f*n_prev + i*k ; ik = i*k ; denom = max(|n_t . q|, 1)
// ---------------------------------------------------------------------------
__global__ __launch_bounds__(256)
void mlstm_gate_vec(const float* __restrict__ n_prev,
                    const float* __restrict__ gates,
                    float* __restrict__ n_t_out,
                    float* __restrict__ ik,
                    float* __restrict__ denom) {
  __shared__ float red[256];
  const float* q  = gates;
  const float* k  = gates + HID;
  const float* ig = gates + 3 * HID;
  const float* f  = gates + 4 * HID;
  float part = 0.0f;
  for (int e = threadIdx.x; e < HID; e += 256) {
    float ikv = ig[e] * k[e];
    float nt  = f[e] * n_prev[e] + ikv;
    ik[e] = ikv;
    n_t_out[e] = nt;
    part += nt * q[e];
  }
  red[threadIdx.x] = part;
  __syncthreads();
  for (int s = 128; s > 0; s >>= 1) {
    if (threadIdx.x < s) red[threadIdx.x] += red[threadIdx.x + s];
    __syncthreads();
  }
  if (threadIdx.x == 0) denom[0] = fmaxf(fabsf(red[0]), 1.0f);
}

// ---------------------------------------------------------------------------
// Kernel 3: one block per row r.
//   c_t[r][c] = f[c]*c_prev[r][c] + v[r]*(i[c]*k[c])   (trailing-dim bcast)
//   h_t[r]    = o[r] * (sum_c c_t[r][c]*q[c]) / denom
// Streams c_prev in / c_t out nontemporally; row dot fused in-register.
// ---------------------------------------------------------------------------
__global__ __launch_bounds__(256)
void mlstm_ct_h(const float* __restrict__ c_prev,
                const float* __restrict__ gates,
                const float* __restrict__ ik,
                const float* __restrict__ denom,
                float* __restrict__ c_t,
                float* __restrict__ h_t) {
  const int r = blockIdx.x;
  __shared__ float red[256];
  const float vr = gates[2 * HID + r];
  const v4f* __restrict__ cp  = (const v4f*)(c_prev + (size_t)r * HID);
  v4f* __restrict__       co  = (v4f*)(c_t + (size_t)r * HID);
  const v4f* __restrict__ f4  = (const v4f*)(gates + 4 * HID);
  const v4f* __restrict__ ik4 = (const v4f*)ik;
  const v4f* __restrict__ q4  = (const v4f*)gates;
  float acc = 0.0f;
  for (int c = threadIdx.x; c < HID / 4; c += 256) {
    v4f cv = __builtin_nontemporal_load(&cp[c]);
    v4f ct = f4[c] * cv + vr * ik4[c];
    __builtin_nontemporal_store(ct, &co[c]);
    v4f p = ct * q4[c];
    acc += (p[0] + p[1]) + (p[2] + p[3]);
  }
  red[threadIdx.x] = acc;
  __syncthreads();
  for (int s = 128; s > 0; s >>= 1) {
    if (threadIdx.x < s) red[threadIdx.x] += red[threadIdx.x + s];
    __syncthreads();
  }
  if (threadIdx.x == 0)
    h_t[r] = gates[5 * HID + r] * red[0] / denom[0];
}

extern "C" void kernel_launch(void* const* d_in, const int* in_sizes, int n_in,
                              void* d_out, int out_size, void* d_ws, size_t ws_size,
                              hipStream_t stream) {
  (void)in_sizes; (void)n_in; (void)out_size; (void)ws_size;
  const float* x      = (const float*)d_in[0];
  // d_in[1] = h_prev (unused by reference)
  const float* c_prev = (const float*)d_in[2];
  const float* n_prev = (const float*)d_in[3];
  const float* Ww[6]; const float* Wb[6];
  for (int j = 0; j < 6; ++j) {
    Ww[j] = (const float*)d_in[4 + 2 * j];
    Wb[j] = (const float*)d_in[5 + 2 * j];
  }

  float* out  = (float*)d_out;
  float* h_t  = out;                                   // [HID]
  float* c_t  = out + HID;                             // [HID*HID]
  float* n_t  = out + HID + (size_t)HID * HID;         // [HID]

  float* gates = (float*)d_ws;                         // 6*HID
  float* ik    = gates + 6 * HID;                      // HID
  float* denom = ik + HID;                             // 1

  mlstm_gemv6<<<1536, 256, 0, stream>>>(x,
      Ww[0], Wb[0], Ww[1], Wb[1], Ww[2], Wb[2],
      Ww[3], Wb[3], Ww[4], Wb[4], Ww[5], Wb[5], gates);
  mlstm_gate_vec<<<1, 256, 0, stream>>>(n_prev, gates, n_t, ik, denom);
  mlstm_ct_h<<<HID, 256, 0, stream>>>(c_prev, gates, ik, denom, c_t, h_t);
}